// ConstraintOptimizer_77841987273011
// MI455X (gfx1250) — compile-verified
//
#include <hip/hip_runtime.h>
#include <hip/hip_bf16.h>
#include <stdint.h>

// ---------------------------------------------------------------------------
// Arc-length trajectory projection onto road polylines (MI455X / gfx1250).
//
// No dense matmul exists in this workload (3-vector geometry + sqrt + binary
// search), so WMMA is not applicable. The CDNA5-specific win is the Tensor
// Data Mover: each candidate consumes a contiguous 256x3 f32 tile (3 KB) of
// road points, which we DMA into LDS with tensor_load_to_lds and overlap with
// mask popcount / trajectory arc-length setup, then fence with
// s_wait_tensorcnt. Wave-private LDS phases are fenced with s_wait_dscnt
// (wave32 lockstep + in-order DS per wave).
// ---------------------------------------------------------------------------

#define N_    1024
#define NB_   16
#define NP_   256
#define T_    80
#define NCAND (NB_ * 2)
#define EPS_LEN 1e-9f
#define EPS_DD  1e-12f

typedef uint32_t v4u __attribute__((ext_vector_type(4)));
typedef int      v4i __attribute__((ext_vector_type(4)));
typedef int      v8i __attribute__((ext_vector_type(8)));

#if __has_include(<hip/amd_detail/amd_gfx1250_TDM.h>)
#define TDM_6ARG 1
#endif

// Wave-level LDS producer->consumer fence: DS ops from one wave are in order,
// but make the dependency explicit to the scheduler/compiler.
__device__ __forceinline__ void wave_lds_fence() {
  __builtin_amdgcn_wave_barrier();
  asm volatile("s_wait_dscnt 0" ::: "memory");
  __builtin_amdgcn_wave_barrier();
}

// 1-D TDM DMA: n_elems f32 from global -> LDS. Descriptor per CDNA5 ISA ch.8.
__device__ __forceinline__ void tdm_load_f32_tile(const float* gsrc,
                                                  uint32_t lds_byte_addr,
                                                  uint32_t n_elems) {
  uint64_t ga = (uint64_t)(uintptr_t)gsrc;
  v4u g0;
  g0[0] = 1u;                                   // count=1, user descriptor
  g0[1] = lds_byte_addr;                        // lds_addr
  g0[2] = (uint32_t)(ga & 0xFFFFFFFFu);         // global_addr[31:0]
  g0[3] = ((uint32_t)(ga >> 32) & 0x01FFFFFFu)  // global_addr[56:32]
          | (2u << 30);                         // type = 2 ("image")
  v8i g1;
  g1[0] = (int)(2u << 16);                      // data_size=2 (4B), mask=0
  g1[1] = (int)((n_elems & 0xFFFFu) << 16);     // tensor_dim0[15:0]
  g1[2] = (int)(((n_elems >> 16) & 0xFFFFu)     // tensor_dim0[31:16]
          | (1u << 16));                        // tensor_dim1 = 1
  g1[3] = (int)((n_elems & 0xFFFFu) << 16);     // tile_dim0
  g1[4] = 1;                                    // tile_dim1 = 1
  g1[5] = (int)n_elems;                         // tensor_dim0_stride lo32
  g1[6] = 0;                                    // stride hi / dim1_stride lo
  g1[7] = 0;
  v4i gz = {0, 0, 0, 0};
#ifdef TDM_6ARG
  v8i gz8 = {0, 0, 0, 0, 0, 0, 0, 0};
  __builtin_amdgcn_tensor_load_to_lds(g0, g1, gz, gz, gz8, 0);
#else
  __builtin_amdgcn_tensor_load_to_lds(g0, g1, gz, gz, 0);
#endif
}

// Point accessor with direction mapping (mask is a length-prefix, so the
// packed-forward sequence is identity order and backward is the reverse).
__device__ __forceinline__ float pt_at(const float* pts, int nv, int dir,
                                       int k, int c) {
  int idx = dir ? (nv - 1 - k) : k;
  return pts[idx * 3 + c];
}

// Wave-cooperative evaluation of one (n, nb, dir) candidate. Requires nv>=2.
// Returns summed cost (uniform across lanes); optionally writes proj[T*3].
__device__ float eval_candidate(int lane, int nv, int dir,
                                const float* __restrict__ pts,  // LDS 256*3
                                float* __restrict__ seg,        // LDS 256
                                float* __restrict__ cum,        // LDS 256
                                const float* __restrict__ pos,  // LDS T*3
                                const float* __restrict__ tcs,  // LDS T
                                float* __restrict__ proj_out) { // global/null
  const int nseg = nv - 1;

  // --- segment lengths (clamped; zero beyond valid range) ---
  for (int i = lane; i < NP_; i += 32) {
    float l = 0.0f;
    if (i < nseg) {
      float dx = pt_at(pts, nv, dir, i + 1, 0) - pt_at(pts, nv, dir, i, 0);
      float dy = pt_at(pts, nv, dir, i + 1, 1) - pt_at(pts, nv, dir, i, 1);
      float dz = pt_at(pts, nv, dir, i + 1, 2) - pt_at(pts, nv, dir, i, 2);
      l = fmaxf(sqrtf(dx * dx + dy * dy + dz * dz), EPS_LEN);
    }
    seg[i] = l;
  }
  wave_lds_fence();

  // --- 256-entry exclusive scan: 8 contiguous per lane + shfl_up of sums ---
  const int base = lane * 8;
  float lsum = 0.0f;
#pragma unroll
  for (int k = 0; k < 8; ++k) lsum += seg[base + k];
  float inc = lsum;
#pragma unroll
  for (int off = 1; off < 32; off <<= 1) {
    float y = __shfl_up(inc, off, 32);
    if (lane >= off) inc += y;
  }
  float run = inc - lsum;  // exclusive prefix of this lane's chunk
#pragma unroll
  for (int k = 0; k < 8; ++k) {
    cum[base + k] = run;   // cum[i] = sum seg[0..i-1]; constant past nseg
    run += seg[base + k];
  }
  wave_lds_fence();
  const float total_s = cum[nseg];

  // --- entry projection of pos[0] onto segments; argmin d2 (first index) ---
  const float p0x = pos[0], p0y = pos[1], p0z = pos[2];
  float bd2 = __builtin_inff(), bent = 0.0f;
  int bidx = 0x7fffffff;
  for (int i = lane; i < nseg; i += 32) {
    float ax = pt_at(pts, nv, dir, i, 0);
    float ay = pt_at(pts, nv, dir, i, 1);
    float az = pt_at(pts, nv, dir, i, 2);
    float vx = pt_at(pts, nv, dir, i + 1, 0) - ax;
    float vy = pt_at(pts, nv, dir, i + 1, 1) - ay;
    float vz = pt_at(pts, nv, dir, i + 1, 2) - az;
    float dd = fmaxf(vx * vx + vy * vy + vz * vz, EPS_DD);
    float tt = ((p0x - ax) * vx + (p0y - ay) * vy + (p0z - az) * vz) / dd;
    tt = fminf(fmaxf(tt, 0.0f), 1.0f);
    float qx = ax + tt * vx, qy = ay + tt * vy, qz = az + tt * vz;
    float ex = p0x - qx, ey = p0y - qy, ez = p0z - qz;
    float d2 = ex * ex + ey * ey + ez * ez;
    if (d2 < bd2 || (d2 == bd2 && i < bidx)) {
      bd2 = d2;
      bent = cum[i] + tt * seg[i];
      bidx = i;
    }
  }
#pragma unroll
  for (int off = 16; off; off >>= 1) {
    float od2 = __shfl_xor(bd2, off, 32);
    float oen = __shfl_xor(bent, off, 32);
    int   oix = __shfl_xor(bidx, off, 32);
    if (od2 < bd2 || (od2 == bd2 && oix < bidx)) {
      bd2 = od2; bent = oen; bidx = oix;
    }
  }
  const float entry_s = bent;

  // --- 80 lane-parallel arc-length lookups + lerp + cost ---
  float cost = 0.0f;
  for (int t = lane; t < T_; t += 32) {
    float tg = fminf(fmaxf(entry_s + tcs[t], 0.0f), total_s);
    int lo = 0, hi = NP_;                 // upper_bound over cum[0..255]
    while (lo < hi) {
      int mid = (lo + hi) >> 1;
      if (cum[mid] <= tg) lo = mid + 1; else hi = mid;
    }
    int j = lo - 1;
    j = j < 0 ? 0 : j;
    j = j > nv - 2 ? nv - 2 : j;
    float tl = (tg - cum[j]) / fmaxf(seg[j], EPS_LEN);
    tl = fminf(fmaxf(tl, 0.0f), 1.0f);
    float pax = pt_at(pts, nv, dir, j, 0);
    float pay = pt_at(pts, nv, dir, j, 1);
    float paz = pt_at(pts, nv, dir, j, 2);
    float px = pax + tl * (pt_at(pts, nv, dir, j + 1, 0) - pax);
    float py = pay + tl * (pt_at(pts, nv, dir, j + 1, 1) - pay);
    float pz = paz + tl * (pt_at(pts, nv, dir, j + 1, 2) - paz);
    float dx = pos[t * 3 + 0] - px;
    float dy = pos[t * 3 + 1] - py;
    float dz = pos[t * 3 + 2] - pz;
    cost += sqrtf(dx * dx + dy * dy + dz * dz);
    if (proj_out) {
      proj_out[t * 3 + 0] = px;
      proj_out[t * 3 + 1] = py;
      proj_out[t * 3 + 2] = pz;
    }
  }
#pragma unroll
  for (int off = 16; off; off >>= 1) cost += __shfl_xor(cost, off, 32);
  return cost;
}

// Per-block setup shared by both kernels: TDM the point tile, popcount the
// mask, stage trajectory positions + arc-length prefix. Returns n_valid.
// Leaves TENSORcnt waited and LDS fenced.
__device__ int setup_candidate(int lane, int n, int nb,
                               const float* __restrict__ traj,
                               const float* __restrict__ rp,
                               const unsigned char* __restrict__ rm,
                               float* s_pts, float* s_pos, float* s_tcs,
                               float* s_d) {
  const float* rp_tile = rp + (size_t)(n * NB_ + nb) * NP_ * 3;
  const unsigned char* m = rm + (size_t)(n * NB_ + nb) * NP_;
  __builtin_prefetch(m, 0, 0);  // global_prefetch_b8

  // Kick off the DMA first; overlap with the setup below.
  tdm_load_f32_tile(rp_tile, (uint32_t)(uintptr_t)(void*)s_pts, NP_ * 3);

  // n_valid = popcount of mask bytes (8 bytes per lane).
  const unsigned long long* m8 = (const unsigned long long*)m;
  unsigned long long mv = m8[lane];
  int cnt = __popcll(mv & 0x0101010101010101ULL);
#pragma unroll
  for (int off = 16; off; off >>= 1) cnt += __shfl_xor(cnt, off, 32);

  // Stage trajectory positions.
  const float* pr = traj + (size_t)n * T_ * 3;
  for (int t = lane; t < T_; t += 32) {
    s_pos[t * 3 + 0] = pr[t * 3 + 0];
    s_pos[t * 3 + 1] = pr[t * 3 + 1];
    s_pos[t * 3 + 2] = pr[t * 3 + 2];
  }
  wave_lds_fence();

  // Trajectory segment distances, then a short serial scan on lane 0.
  for (int t = lane; t < T_; t += 32) {
    float d = 0.0f;
    if (t >= 1) {
      float dx = s_pos[t * 3 + 0] - s_pos[(t - 1) * 3 + 0];
      float dy = s_pos[t * 3 + 1] - s_pos[(t - 1) * 3 + 1];
      float dz = s_pos[t * 3 + 2] - s_pos[(t - 1) * 3 + 2];
      d = sqrtf(dx * dx + dy * dy + dz * dz);
    }
    s_d[t] = d;
  }
  wave_lds_fence();
  if (lane == 0) {
    float a = 0.0f;
    for (int t = 0; t < T_; ++t) { a += s_d[t]; s_tcs[t] = a; }
  }
  wave_lds_fence();

  // Point tile must be resident before eval reads it.
  __builtin_amdgcn_s_wait_tensorcnt(0);
  __builtin_amdgcn_wave_barrier();
  return cnt;
}

// Kernel 1: one wave32 block per (n, nb, dir) candidate -> cost to workspace.
__global__ __launch_bounds__(32) void k_candidate_cost(
    const float* __restrict__ traj, const float* __restrict__ rp,
    const unsigned char* __restrict__ rm, float* __restrict__ cost_ws) {
  __shared__ __align__(16) float s_pts[NP_ * 3];
  __shared__ float s_seg[NP_];
  __shared__ float s_cum[NP_];
  __shared__ float s_pos[T_ * 3];
  __shared__ float s_tcs[T_];
  __shared__ float s_d[T_];

  const int b = blockIdx.x;
  const int n = b / NCAND;
  const int r = b - n * NCAND;
  const int nb = r >> 1, dir = r & 1;
  const int lane = threadIdx.x;

  int nv = setup_candidate(lane, n, nb, traj, rp, rm, s_pts, s_pos, s_tcs, s_d);

  float c = __builtin_inff();
  if (nv >= 2)
    c = eval_candidate(lane, nv, dir, s_pts, s_seg, s_cum, s_pos, s_tcs,
                       nullptr);
  if (lane == 0) cost_ws[b] = c;
}

// Kernel 2: one wave per trajectory: argmin over 32 candidate costs (lower
// index wins ties, matching jnp.argmin), then recompute only the winner's
// projection and write it out. All-invalid -> pass-through positions.
__global__ __launch_bounds__(32) void k_select_and_project(
    const float* __restrict__ traj, const float* __restrict__ rp,
    const unsigned char* __restrict__ rm, const float* __restrict__ cost_ws,
    float* __restrict__ out) {
  __shared__ __align__(16) float s_pts[NP_ * 3];
  __shared__ float s_seg[NP_];
  __shared__ float s_cum[NP_];
  __shared__ float s_pos[T_ * 3];
  __shared__ float s_tcs[T_];
  __shared__ float s_d[T_];

  const int n = blockIdx.x;
  const int lane = threadIdx.x;

  float c = cost_ws[(size_t)n * NCAND + lane];
  int idx = lane;
#pragma unroll
  for (int off = 16; off; off >>= 1) {
    float oc = __shfl_xor(c, off, 32);
    int oi = __shfl_xor(idx, off, 32);
    if (oc < c || (oc == c && oi < idx)) { c = oc; idx = oi; }
  }

  const float* pr = traj + (size_t)n * T_ * 3;
  float* po = out + (size_t)n * T_ * 3;

  if (!(c < __builtin_inff())) {  // no candidate with n_valid >= 2
    for (int t = lane; t < T_; t += 32) {
      po[t * 3 + 0] = pr[t * 3 + 0];
      po[t * 3 + 1] = pr[t * 3 + 1];
      po[t * 3 + 2] = pr[t * 3 + 2];
    }
    return;
  }

  const int nb = idx >> 1, dir = idx & 1;
  int nv = setup_candidate(lane, n, nb, traj, rp, rm, s_pts, s_pos, s_tcs, s_d);
  (void)eval_candidate(lane, nv, dir, s_pts, s_seg, s_cum, s_pos, s_tcs, po);
}

extern "C" void kernel_launch(void* const* d_in, const int* in_sizes, int n_in,
                              void* d_out, int out_size, void* d_ws,
                              size_t ws_size, hipStream_t stream) {
  const float* traj = (const float*)d_in[0];          // (N, T, 3) f32
  const float* rp = (const float*)d_in[1];            // (N, NB, NP, 3) f32
  const unsigned char* rm = (const unsigned char*)d_in[2];  // (N, NB, NP) bool
  float* cost_ws = (float*)d_ws;                      // N*NB*2 floats = 128 KB
  float* out = (float*)d_out;                         // (N, T, 3) f32

  k_candidate_cost<<<N_ * NCAND, 32, 0, stream>>>(traj, rp, rm, cost_ws);
  k_select_and_project<<<N_, 32, 0, stream>>>(traj, rp, rm, cost_ws, out);
}